// RPNLoss_80109730005424
// MI455X (gfx1250) — compile-verified
//
#include <hip/hip_runtime.h>
#include <math.h>

#define B_ 4
#define N_ 200000
#define G_ 64
#define BLK 256
#define NB ((N_ + BLK - 1) / BLK)   /* 782 */
#define MAX_POS 128
#define BATCH_PER_IMG 256
#define FG_THRESH 0.7f
#define BG_THRESH 0.3f

typedef __attribute__((ext_vector_type(2))) float v2f;
typedef __attribute__((ext_vector_type(8))) float v8f;

// ---------------------------------------------------------------------------
// K1: per-anchor max IoU + first-argmax over 64 GT boxes; pack result byte;
//     per-block pos/neg counts.
// ---------------------------------------------------------------------------
__global__ __launch_bounds__(BLK) void k_iou(
    const float* __restrict__ anchors, const float* __restrict__ gt,
    unsigned char* __restrict__ flags,
    int* __restrict__ blk_pos, int* __restrict__ blk_neg)
{
  __shared__ float sg[G_ * 4];
  __shared__ int s_pos, s_neg;
  const int b = blockIdx.y;
  const int t = threadIdx.x;
  const int i = blockIdx.x * BLK + t;
  if (t == 0) { s_pos = 0; s_neg = 0; }
  sg[t] = gt[b * G_ * 4 + t];           // 64 boxes * 4 floats = 256 loads
  __syncthreads();

  int pos = 0, neg = 0;
  if (i < N_) {
    const float4 a = ((const float4*)anchors)[b * N_ + i];
    const float a_area = (a.z - a.x) * (a.w - a.y);
    float best = -1.0f; int bidx = 0;
#pragma unroll 8
    for (int g = 0; g < G_; ++g) {
      const float gx1 = sg[g*4+0], gy1 = sg[g*4+1];
      const float gx2 = sg[g*4+2], gy2 = sg[g*4+3];
      const float g_area = (gx2 - gx1) * (gy2 - gy1);
      const float lx = fmaxf(a.x, gx1), ly = fmaxf(a.y, gy1);
      const float rx = fminf(a.z, gx2), ry = fminf(a.w, gy2);
      const float w = fmaxf(rx - lx, 0.0f), h = fmaxf(ry - ly, 0.0f);
      const float inter = w * h;
      const float iou = inter / (a_area + g_area - inter);
      if (iou > best) { best = iou; bidx = g; }   // strict > == first argmax
    }
    pos = (best >= FG_THRESH) ? 1 : 0;
    neg = (best <  BG_THRESH) ? 1 : 0;
    flags[b * N_ + i] = (unsigned char)(bidx | (pos << 6) | (neg << 7));
  }
  if (pos) atomicAdd(&s_pos, 1);
  if (neg) atomicAdd(&s_neg, 1);
  __syncthreads();
  if (t == 0) {
    blk_pos[b * NB + blockIdx.x] = s_pos;
    blk_neg[b * NB + blockIdx.x] = s_neg;
  }
}

// ---------------------------------------------------------------------------
// K2: per-image scan of 782 block counts (packed u64), exclusive offsets,
//     clamped selection counts, and zero the sample slots.
// ---------------------------------------------------------------------------
__global__ __launch_bounds__(1024) void k_scan(
    const int* __restrict__ blk_pos, const int* __restrict__ blk_neg,
    int* __restrict__ off_pos, int* __restrict__ off_neg,
    int* __restrict__ sel_counts,
    float* __restrict__ slot_bce, float* __restrict__ slot_sl1)
{
  __shared__ unsigned long long sd[1024];
  const int b = blockIdx.x;
  const int t = threadIdx.x;
  int cp = 0, cn = 0;
  if (t < NB) { cp = blk_pos[b * NB + t]; cn = blk_neg[b * NB + t]; }
  const unsigned long long v =
      ((unsigned long long)(unsigned)cp << 32) | (unsigned)cn;
  sd[t] = v;
  __syncthreads();
  for (int off = 1; off < 1024; off <<= 1) {
    unsigned long long add = (t >= off) ? sd[t - off] : 0ull;
    __syncthreads();
    sd[t] += add;
    __syncthreads();
  }
  const unsigned long long incl = sd[t];
  if (t < NB) {
    const unsigned long long excl = incl - v;   // halves never borrow
    off_pos[b * NB + t] = (int)(excl >> 32);
    off_neg[b * NB + t] = (int)(excl & 0xffffffffull);
  }
  if (t == 0) {
    const unsigned long long tot = sd[1023];
    const int tp = (int)(tot >> 32);
    const int tn = (int)(tot & 0xffffffffull);
    const int np = tp < MAX_POS ? tp : MAX_POS;
    const int cap = BATCH_PER_IMG - np;
    const int nn = tn < cap ? tn : cap;
    sel_counts[b * 2 + 0] = np;
    sel_counts[b * 2 + 1] = nn;
  }
  if (t < BATCH_PER_IMG) {
    slot_bce[b * BATCH_PER_IMG + t] = 0.0f;
    slot_sl1[b * BATCH_PER_IMG + t] = 0.0f;
  }
}

// ---------------------------------------------------------------------------
// K3: rank each pos/neg anchor (block offset + intra-block scan); selected
//     anchors write BCE (and smooth-L1 for positives) into fixed slots.
// ---------------------------------------------------------------------------
__global__ __launch_bounds__(BLK) void k_select(
    const float* __restrict__ logits, const float* __restrict__ breg,
    const float* __restrict__ anchors, const float* __restrict__ gt,
    const unsigned char* __restrict__ flags,
    const int* __restrict__ off_pos, const int* __restrict__ off_neg,
    const int* __restrict__ sel_counts,
    float* __restrict__ slot_bce, float* __restrict__ slot_sl1)
{
  __shared__ int sd[BLK];
  const int b = blockIdx.y;
  const int t = threadIdx.x;
  const int i = blockIdx.x * BLK + t;
  const unsigned f = (i < N_) ? (unsigned)flags[b * N_ + i] : 0u;
  const int pos = (f >> 6) & 1;
  const int neg = (f >> 7) & 1;
  const int gidx = (int)(f & 63u);
  const int v = (pos << 16) | neg;      // counts <= 256, no cross-carry
  sd[t] = v;
  __syncthreads();
  for (int off = 1; off < BLK; off <<= 1) {
    int add = (t >= off) ? sd[t - off] : 0;
    __syncthreads();
    sd[t] += add;
    __syncthreads();
  }
  const int incl = sd[t];
  const int npsel = sel_counts[b * 2 + 0];
  const int nnsel = sel_counts[b * 2 + 1];

  if (pos) {
    const int rank = off_pos[b * NB + blockIdx.x] + ((incl >> 16) - pos);
    if (rank < npsel) {
      const float l = logits[b * N_ + i];
      const float bce = fmaxf(l, 0.0f) - l + log1pf(expf(-fabsf(l)));
      const float4 a  = ((const float4*)anchors)[b * N_ + i];
      const float4 tb = ((const float4*)gt)[b * G_ + gidx];
      const float aw = a.z - a.x, ah = a.w - a.y;
      const float acx = (a.x + a.z) * 0.5f, acy = (a.y + a.w) * 0.5f;
      const float tw = tb.z - tb.x, th = tb.w - tb.y;
      const float tcx = (tb.x + tb.z) * 0.5f, tcy = (tb.y + tb.w) * 0.5f;
      const float d0 = (tcx - acx) / aw;
      const float d1 = (tcy - acy) / ah;
      const float d2 = logf(tw / aw);
      const float d3 = logf(th / ah);
      const float4 r = ((const float4*)breg)[b * N_ + i];
      float s = 0.0f, df;
      df = fabsf(r.x - d0); s += (df < 1.0f) ? 0.5f * df * df : df - 0.5f;
      df = fabsf(r.y - d1); s += (df < 1.0f) ? 0.5f * df * df : df - 0.5f;
      df = fabsf(r.z - d2); s += (df < 1.0f) ? 0.5f * df * df : df - 0.5f;
      df = fabsf(r.w - d3); s += (df < 1.0f) ? 0.5f * df * df : df - 0.5f;
      slot_bce[b * BATCH_PER_IMG + rank] = bce;
      slot_sl1[b * BATCH_PER_IMG + rank] = s;
    }
  } else if (neg) {
    const int rank = off_neg[b * NB + blockIdx.x] + ((incl & 0xffff) - neg);
    if (rank < nnsel) {
      const float l = logits[b * N_ + i];
      slot_bce[b * BATCH_PER_IMG + npsel + rank] =
          fmaxf(l, 0.0f) + log1pf(expf(-fabsf(l)));
    }
  }
}

// ---------------------------------------------------------------------------
// K4: single wave32. Deterministic fixed-order reduction of the 1024 slots
//     via chained V_WMMA_F32_16X16X4_F32 against an all-ones B matrix.
//     Sum of all 256 D elements == 16 * grand total (layout-invariant).
// ---------------------------------------------------------------------------
__global__ __launch_bounds__(32) void k_finalize(
    const float* __restrict__ slot_bce, const float* __restrict__ slot_sl1,
    const int* __restrict__ sel_counts, float* __restrict__ out)
{
  const int L = threadIdx.x;
  v2f ones = {1.0f, 1.0f};
  v8f acc_b = {};
  v8f acc_s = {};
#pragma unroll
  for (int c = 0; c < 16; ++c) {        // 16 chunks * 64 values = 1024 slots
    const float* pb = slot_bce + c * 64;
    const float* ps = slot_sl1 + c * 64;
    v2f ab, as;
    ab.x = pb[2 * L]; ab.y = pb[2 * L + 1];   // 64 distinct A elements/chunk
    as.x = ps[2 * L]; as.y = ps[2 * L + 1];
    acc_b = __builtin_amdgcn_wmma_f32_16x16x4_f32(
        false, ab, false, ones, (short)0, acc_b, false, false);
    acc_s = __builtin_amdgcn_wmma_f32_16x16x4_f32(
        false, as, false, ones, (short)0, acc_s, false, false);
  }
  float sb = acc_b[0]+acc_b[1]+acc_b[2]+acc_b[3]+acc_b[4]+acc_b[5]+acc_b[6]+acc_b[7];
  float ss = acc_s[0]+acc_s[1]+acc_s[2]+acc_s[3]+acc_s[4]+acc_s[5]+acc_s[6]+acc_s[7];
#pragma unroll
  for (int off = 16; off >= 1; off >>= 1) {
    sb += __shfl_xor(sb, off, 32);
    ss += __shfl_xor(ss, off, 32);
  }
  sb *= (1.0f / 16.0f);                 // D replicates totals across 16 cols
  ss *= (1.0f / 16.0f);
  if (L == 0) {
    int valid = 0, npos = 0;
    for (int b = 0; b < B_; ++b) {
      const int np = sel_counts[2 * b + 0];
      const int nn = sel_counts[2 * b + 1];
      valid += np + nn;
      npos  += np;
    }
    out[0] = sb / fmaxf((float)valid, 1.0f);
    out[1] = ss / fmaxf((float)npos * 4.0f, 1.0f);
  }
}

// ---------------------------------------------------------------------------
extern "C" void kernel_launch(void* const* d_in, const int* in_sizes, int n_in,
                              void* d_out, int out_size, void* d_ws, size_t ws_size,
                              hipStream_t stream) {
  (void)in_sizes; (void)n_in; (void)out_size; (void)ws_size;
  const float* logits  = (const float*)d_in[0];   // [B,N,1]
  const float* breg    = (const float*)d_in[1];   // [B,N,4]
  const float* anchors = (const float*)d_in[2];   // [B,N,4]
  const float* gt      = (const float*)d_in[3];   // [B,G,4]
  float* out = (float*)d_out;

  char* ws = (char*)d_ws;
  size_t o = 0;
  unsigned char* flags = (unsigned char*)(ws + o); o += (size_t)B_ * N_;
  o = (o + 255) & ~(size_t)255;
  int* blk_pos = (int*)(ws + o); o += (size_t)B_ * NB * sizeof(int);
  int* blk_neg = (int*)(ws + o); o += (size_t)B_ * NB * sizeof(int);
  int* off_pos = (int*)(ws + o); o += (size_t)B_ * NB * sizeof(int);
  int* off_neg = (int*)(ws + o); o += (size_t)B_ * NB * sizeof(int);
  int* sel_counts = (int*)(ws + o); o += (size_t)B_ * 2 * sizeof(int);
  o = (o + 255) & ~(size_t)255;
  float* slot_bce = (float*)(ws + o); o += (size_t)B_ * BATCH_PER_IMG * sizeof(float);
  float* slot_sl1 = (float*)(ws + o);

  k_iou<<<dim3(NB, B_), BLK, 0, stream>>>(anchors, gt, flags, blk_pos, blk_neg);
  k_scan<<<dim3(B_), 1024, 0, stream>>>(blk_pos, blk_neg, off_pos, off_neg,
                                        sel_counts, slot_bce, slot_sl1);
  k_select<<<dim3(NB, B_), BLK, 0, stream>>>(logits, breg, anchors, gt, flags,
                                             off_pos, off_neg, sel_counts,
                                             slot_bce, slot_sl1);
  k_finalize<<<dim3(1), 32, 0, stream>>>(slot_bce, slot_sl1, sel_counts, out);
}